// Lattice_21629455303213
// MI455X (gfx1250) — compile-verified
//
#include <hip/hip_runtime.h>

// Shapes: B=64, C=128, U=4, P=Q=G=H=8, D=4
// Per (b,c):  A[64x4]  rows M = p*8+q, K = u   (from mps[b,c,u,p,q])
//             B[4x256] cols n = g*32+h*4+d     (from mpo[b,u,c,g,h,d])
//             out tile 64x256, mem offset = p*2048 + g*256 + q*32 + (h*4+d)

typedef __attribute__((ext_vector_type(2))) float v2f;
typedef __attribute__((ext_vector_type(8))) float v8f;

__global__ __launch_bounds__(256)
void mps_mpo_contract_wmma(const float* __restrict__ mps,
                           const float* __restrict__ mpo,
                           float* __restrict__ out) {
    __shared__ float a_lds[4 * 64];    // [u][M],  M = p*8+q
    __shared__ float b_lds[4 * 256];   // [u][n],  n = g*32+h*4+d

    const int tid = threadIdx.x;
    const int bc  = blockIdx.x;        // b*128 + c
    const int b   = bc >> 7;
    const int c   = bc & 127;

    // Stage A: mps[b,c,:,:,:] = 256 contiguous floats (flat = u*64 + p*8 + q)
    a_lds[tid] = mps[(size_t)bc * 256 + tid];

    // Stage B: 4 rows of 256 contiguous floats: mpo[b,u,c,:,:,:]
    #pragma unroll
    for (int u = 0; u < 4; ++u) {
        size_t off = ((size_t)(b * 4 + u) * 128 + c) * 256 + tid;
        b_lds[u * 256 + tid] = mpo[off];
    }
    __syncthreads();

    const int wid  = tid >> 5;         // 8 waves
    const int lane = tid & 31;
    const int half = lane >> 4;        // K-half select (0: K=0,1  1: K=2,3)
    const int l16  = lane & 15;

    const int mt  = wid & 3;           // M-tile 0..3 (rows mt*16 .. mt*16+15)
    const int ct0 = (wid >> 2) * 8;    // this wave's 8 N-tiles

    // A fragment: V0 = K = 2*half, V1 = K = 2*half+1, row M = mt*16 + l16
    v2f a;
    a.x = a_lds[(half * 2 + 0) * 64 + mt * 16 + l16];
    a.y = a_lds[(half * 2 + 1) * 64 + mt * 16 + l16];

    float* __restrict__ outbase = out + (size_t)bc * 16384;

    #pragma unroll
    for (int t = 0; t < 8; ++t) {
        const int ct = ct0 + t;
        const int n  = ct * 16 + l16;              // column 0..255
        // B fragment: V0 = K = 2*half, V1 = K = 2*half+1, col N = n
        v2f bf;
        bf.x = b_lds[(half * 2 + 0) * 256 + n];
        bf.y = b_lds[(half * 2 + 1) * 256 + n];

        v8f acc = {};
        acc = __builtin_amdgcn_wmma_f32_16x16x4_f32(
            /*neg_a=*/false, a, /*neg_b=*/false, bf,
            /*c_mod=*/(short)0, acc, /*reuse_a=*/false, /*reuse_b=*/false);

        // Column part of the address: g = n>>5 never changes within a 16-wide tile,
        // so 16 consecutive lanes hit 16 contiguous floats (64B segment).
        const int colOff = (n >> 5) * 256 + (n & 31);

        #pragma unroll
        for (int i = 0; i < 8; ++i) {
            const int Rw = mt * 16 + half * 8 + i; // D VGPR i -> row i (lanes 0-15) / i+8
            const int p  = Rw >> 3;
            const int q  = Rw & 7;
            __builtin_nontemporal_store(acc[i], outbase + p * 2048 + q * 32 + colOff);
        }
    }
}

extern "C" void kernel_launch(void* const* d_in, const int* in_sizes, int n_in,
                              void* d_out, int out_size, void* d_ws, size_t ws_size,
                              hipStream_t stream) {
    const float* mps = (const float*)d_in[0];
    const float* mpo = (const float*)d_in[1];
    float* out = (float*)d_out;

    dim3 grid(64 * 128);   // one block per (b,c)
    dim3 block(256);       // 8 waves
    hipLaunchKernelGGL(mps_mpo_contract_wmma, grid, block, 0, stream, mps, mpo, out);
}